// PredictSpanSet_84739704750407
// MI455X (gfx1250) — compile-verified
//
#include <hip/hip_runtime.h>
#include <hip/hip_bf16.h>
#include <math.h>

// Problem constants (reference: BS=32, BT=1024, DIM=512)
#define BS   32
#define BT   1024
#define DIM  512
#define NEG_MASK (-1000000.0f)
#define NEG_SPAN (-1000.0f)

typedef __attribute__((ext_vector_type(2))) float v2f;
typedef __attribute__((ext_vector_type(4))) float v4f;
typedef __attribute__((ext_vector_type(8))) float v8f;

// ---------------------------------------------------------------------------
// K1: Y[w][b][d] = sum_e x[b,e] * W_w[e,d]   (3 tiny GEMMs, f32 WMMA 16x16x4)
// grid (DIM/16, BS/16, 3), block 32 (one wave per 16x16 C tile)
// ---------------------------------------------------------------------------
__global__ void k1_proj_y(const float* __restrict__ x,
                          const float* __restrict__ Wb,
                          const float* __restrict__ Wi,
                          const float* __restrict__ Wo,
                          float* __restrict__ Y) {
    const int lane = threadIdx.x;
    const int lr = lane & 15;        // N column / M row within half
    const int lh = lane >> 4;        // 0: lanes 0-15, 1: lanes 16-31
    const int d0 = blockIdx.x * 16;
    const int m0 = blockIdx.y * 16;
    const int w  = blockIdx.z;
    const float* W = (w == 0) ? Wb : (w == 1) ? Wi : Wo;

    v8f acc = {};
    for (int k0 = 0; k0 < DIM; k0 += 4) {
        const int ka = k0 + lh * 2;
        // A tile 16x4 of x: lane holds (row=m0+lr, K=ka,ka+1)
        v2f a = *(const v2f*)(x + (m0 + lr) * DIM + ka);
        // B tile 4x16 of W: lane holds (K=ka..ka+1, N=d0+lr)
        v2f bvec;
        bvec.x = W[(ka + 0) * DIM + d0 + lr];
        bvec.y = W[(ka + 1) * DIM + d0 + lr];
        acc = __builtin_amdgcn_wmma_f32_16x16x4_f32(
            false, a, false, bvec, (short)0, acc, false, false);
    }
    // C layout: VGPR j -> M = m0 + j + 8*lh, N = d0 + lr
    #pragma unroll
    for (int j = 0; j < 8; ++j) {
        const int row = m0 + j + lh * 8;  // batch index b
        Y[(w * BS + row) * DIM + d0 + lr] = acc[j];
    }
}

// ---------------------------------------------------------------------------
// K2: bdot[w][b] = bias_w . x[b]    grid (3, BS), block 32 (one wave)
// ---------------------------------------------------------------------------
__global__ void k2_bias_dot(const float* __restrict__ x,
                            const float* __restrict__ bb,
                            const float* __restrict__ bi,
                            const float* __restrict__ bo,
                            float* __restrict__ bdot) {
    const int w = blockIdx.x, b = blockIdx.y, lane = threadIdx.x;
    const float* bias = (w == 0) ? bb : (w == 1) ? bi : bo;
    const float* xr = x + b * DIM;
    float s = 0.f;
    for (int e = lane; e < DIM; e += 32) s += bias[e] * xr[e];
    #pragma unroll
    for (int off = 16; off; off >>= 1) s += __shfl_down(s, off, 32);
    if (lane == 0) bdot[w * BS + b] = s;
}

// ---------------------------------------------------------------------------
// K3: scores + logsumexp.  Per batch b: sen[b](1024x512) @ Ypad(512x16) via
// f32 WMMA 16x16x4; cols 0..2 of C are raw b/i/o scores.
// grid (BT/64, BS), block 128 (4 waves, each owns a 16-row tile)
// ---------------------------------------------------------------------------
__global__ void k3_score_lse(const float* __restrict__ sen,
                             const int*   __restrict__ mask,
                             const float* __restrict__ Y,
                             const float* __restrict__ bdot,
                             float* __restrict__ SB, float* __restrict__ SI,
                             float* __restrict__ SO, float* __restrict__ LSE) {
    __shared__ float ysh[DIM * 16];     // 32 KB: Y[b] padded 512x16 (N<3 live)
    __shared__ float csh[4][16 * 16];   // per-wave C spill

    const int tid  = threadIdx.x;
    const int b    = blockIdx.y;
    const int wave = tid >> 5;
    const int lane = tid & 31;
    const int lr = lane & 15, lh = lane >> 4;

    // Stage zero-padded B operand into LDS (coalesced-ish, once per block)
    for (int i = tid; i < DIM * 16; i += 128) {
        const int d = i >> 4, n = i & 15;
        ysh[i] = (n < 3) ? Y[(n * BS + b) * DIM + d] : 0.f;
    }
    __syncthreads();

    const int m0 = blockIdx.x * 64 + wave * 16;
    const float* srow = sen + ((size_t)b * BT + m0 + lr) * DIM;

    v8f acc = {};
    for (int k0 = 0; k0 < DIM; k0 += 4) {
        const int ka = k0 + lh * 2;
        v2f a = *(const v2f*)(srow + ka);           // A: 16 sen rows x 4 K
        v2f bvec;
        bvec.x = ysh[(ka + 0) * 16 + lr];
        bvec.y = ysh[(ka + 1) * 16 + lr];
        acc = __builtin_amdgcn_wmma_f32_16x16x4_f32(
            false, a, false, bvec, (short)0, acc, false, false);
    }
    #pragma unroll
    for (int j = 0; j < 8; ++j)
        csh[wave][(j + lh * 8) * 16 + lr] = acc[j];
    __syncthreads();

    if (lane < 16) {
        const int r = m0 + lane;
        const float rb = csh[wave][lane * 16 + 0] + bdot[0 * BS + b];
        const float ri = csh[wave][lane * 16 + 1] + bdot[1 * BS + b];
        const float ro = csh[wave][lane * 16 + 2] + bdot[2 * BS + b];
        const int mk = mask[b * BT + r];
        const float sb = mk ? rb : NEG_MASK;
        const float si = mk ? ri : NEG_MASK;
        const float so = mk ? ro : 0.f;
        const float m  = fmaxf(sb, fmaxf(si, so));
        const float lse = m + logf(expf(sb - m) + expf(si - m) + expf(so - m));
        SB[b * BT + r] = sb;  SI[b * BT + r] = si;
        SO[b * BT + r] = so;  LSE[b * BT + r] = lse;
    }
}

// ---------------------------------------------------------------------------
// K4: per-batch inclusive scans of lse & si (Hillis-Steele in LDS), then
//   A[b,t] = sb + lse_excl - si_excl - full
//   B[b,t] = si_excl + (full - lse_incl) + so     (bwd[t+1] == full - lse_incl[t])
// grid BS, block BT
// ---------------------------------------------------------------------------
__global__ void k4_scan(const float* __restrict__ SB, const float* __restrict__ SI,
                        const float* __restrict__ SO, const float* __restrict__ LSE,
                        float* __restrict__ Av, float* __restrict__ Bv) {
    __shared__ float sl[BT];
    __shared__ float ss[BT];
    const int t = threadIdx.x, b = blockIdx.x;
    const float lv = LSE[b * BT + t];
    const float sv = SI[b * BT + t];
    sl[t] = lv; ss[t] = sv;
    __syncthreads();
    for (int off = 1; off < BT; off <<= 1) {
        const float al = (t >= off) ? sl[t - off] : 0.f;
        const float as = (t >= off) ? ss[t - off] : 0.f;
        __syncthreads();
        sl[t] += al; ss[t] += as;
        __syncthreads();
    }
    const float linc = sl[t], sinc = ss[t];
    const float full = sl[BT - 1];
    const float lexc = linc - lv, sexc = sinc - sv;
    Av[b * BT + t] = SB[b * BT + t] + lexc - sexc - full;
    Bv[b * BT + t] = sexc + (full - linc) + SO[b * BT + t];
}

// ---------------------------------------------------------------------------
// K5: span[b,i,j] = (j>=i) ? A[b,i]+B[b,j] : NEG_SPAN   (134 MB stream-out)
// grid BS*BT (one block per output row), block 256, float4 NT stores
// ---------------------------------------------------------------------------
__global__ void k5_span(const float* __restrict__ Av,
                        const float* __restrict__ Bv,
                        float* __restrict__ out) {
    const int r = blockIdx.x;          // r = b*BT + i
    const int b = r >> 10;
    const int i = r & (BT - 1);
    const int j0 = threadIdx.x * 4;
    const float a = Av[r];
    const v4f bv = *(const v4f*)(Bv + (b << 10) + j0);
    v4f o;
    o.x = (j0 + 0 >= i) ? a + bv.x : NEG_SPAN;
    o.y = (j0 + 1 >= i) ? a + bv.y : NEG_SPAN;
    o.z = (j0 + 2 >= i) ? a + bv.z : NEG_SPAN;
    o.w = (j0 + 3 >= i) ? a + bv.w : NEG_SPAN;
    __builtin_nontemporal_store(o, (v4f*)(out + ((size_t)r << 10) + j0));
}

// ---------------------------------------------------------------------------
extern "C" void kernel_launch(void* const* d_in, const int* in_sizes, int n_in,
                              void* d_out, int out_size, void* d_ws, size_t ws_size,
                              hipStream_t stream) {
    const float* x   = (const float*)d_in[0];
    const float* sen = (const float*)d_in[1];
    const int*   msk = (const int*)  d_in[2];
    const float* Wb  = (const float*)d_in[3];
    const float* bb  = (const float*)d_in[4];
    const float* Wi  = (const float*)d_in[5];
    const float* bi  = (const float*)d_in[6];
    const float* Wo  = (const float*)d_in[7];
    const float* bo  = (const float*)d_in[8];
    float* out = (float*)d_out;

    // Workspace layout (floats): ~246K floats (~1 MB)
    float* ws   = (float*)d_ws;
    float* Y    = ws;                      // 3*BS*DIM = 49152
    float* bdot = Y + 3 * BS * DIM;        // 96
    float* SB   = bdot + 3 * BS;           // BS*BT each below
    float* SI   = SB + BS * BT;
    float* SO   = SI + BS * BT;
    float* LSE  = SO + BS * BT;
    float* Av   = LSE + BS * BT;
    float* Bv   = Av + BS * BT;

    k1_proj_y  <<<dim3(DIM / 16, BS / 16, 3), 32,  0, stream>>>(x, Wb, Wi, Wo, Y);
    k2_bias_dot<<<dim3(3, BS),               32,  0, stream>>>(x, bb, bi, bo, bdot);
    k3_score_lse<<<dim3(BT / 64, BS),        128, 0, stream>>>(sen, msk, Y, bdot,
                                                               SB, SI, SO, LSE);
    k4_scan    <<<BS,                        BT,  0, stream>>>(SB, SI, SO, LSE, Av, Bv);
    k5_span    <<<BS * BT,                   256, 0, stream>>>(Av, Bv, out);
}